// PhysicsLoss_17059610100485
// MI455X (gfx1250) — compile-verified
//
#include <hip/hip_runtime.h>
#include <hip/hip_bf16.h>
#include <stdint.h>

// ---------------------------------------------------------------------------
// PhysicsLoss for MI455X (gfx1250, wave32).
// Fused streaming pass + LDS-aggregated segment sums + histogram radix-select
// quantile (async-to-LDS staged) + WMMA ones-matmul partials reduction +
// f64 max-plus scan for the queue recurrence.
// ---------------------------------------------------------------------------

#define N_ROWS   4000000
#define NWIN     4096
#define ALPHA_C  0.1f
#define BETA_C   0.1f
#define CAPACITY 1.0e9f
#define DELTA_T  0.1f
#define SERVICE  1.0e8f      // CAPACITY * DELTA_T (exact in f32)
#define PBLK     256         // blocks in main pass (partials sized for this)
#define TMAIN    512
#define TILE_KEYS 1024       // keys per async-staged tile (256 thr x 4)

typedef float v2f __attribute__((ext_vector_type(2)));
typedef float v8f __attribute__((ext_vector_type(8)));

struct SelState {
    unsigned prefix[2];   // accumulating bit patterns of xs[lo], xs[hi]
    unsigned rank[2];     // residual rank within current prefix
    float    frac;
    float    xs0, xs1;
    float    ref_dobs;
    float    n_valid;
    unsigned pad[3];
};

// --- CDNA5 async global->LDS helpers ---------------------------------------
__device__ __forceinline__ unsigned lds_off_u32(const void* p) {
    // generic pointer to LDS: low 32 bits are the workgroup-relative LDS addr
    return (unsigned)(uintptr_t)p;
}
__device__ __forceinline__ void async_ld_b128(unsigned ldsoff, const unsigned* g) {
    asm volatile("global_load_async_to_lds_b128 %0, %1, off"
                 :: "v"(ldsoff), "v"(g) : "memory");
}
__device__ __forceinline__ void wait_async_le1() {
    asm volatile("s_wait_asynccnt 0x1" ::: "memory");
}
__device__ __forceinline__ void wait_async_le0() {
    asm volatile("s_wait_asynccnt 0x0" ::: "memory");
}

// ---------------------------------------------------------------------------
__global__ void zero_kernel(unsigned* __restrict__ h0g, unsigned* __restrict__ rhg,
                            SelState* st, double* __restrict__ dacc) {
    const int tid = threadIdx.x;
    for (int j = tid; j < 2048; j += 256) h0g[j] = 0u;
    for (int j = tid; j < 512;  j += 256) rhg[j] = 0u;
    if (tid == 0) {
        st->prefix[0] = 0u; st->prefix[1] = 0u;
        st->rank[0] = 0u;   st->rank[1] = 0u;
        st->frac = 0.f; st->xs0 = 0.f; st->xs1 = 0.f;
        st->ref_dobs = 1e-6f; st->n_valid = 0.f;
        dacc[0] = 0.0; dacc[1] = 0.0;
    }
}

// ---------------------------------------------------------------------------
// Main streaming pass: cross-entropy partials (f64 atomics), per-window LDS
// segment sums (ds_add_f32), quantile key emission.
// ---------------------------------------------------------------------------
__global__ __launch_bounds__(TMAIN) void main_kernel(
    const float* __restrict__ logits, const int* __restrict__ y,
    const int* __restrict__ mask, const float* __restrict__ xraw,
    const int* __restrict__ widx, const float* __restrict__ cw,
    unsigned* __restrict__ keys, float* __restrict__ partials,
    double* __restrict__ dacc)
{
    __shared__ float wsum[4 * NWIN];   // 64 KB of the 320 KB WGP LDS
    for (int j = threadIdx.x; j < 4 * NWIN; j += blockDim.x) wsum[j] = 0.f;
    __syncthreads();

    const float cw0 = cw[0], cw1 = cw[1];
    double a_wnll = 0.0, a_w = 0.0;
    const int stride = gridDim.x * blockDim.x;

    for (int i = blockIdx.x * blockDim.x + threadIdx.x; i < N_ROWS; i += stride) {
        __builtin_prefetch(xraw + 8 * (i + stride), 0, 0);   // global_prefetch_b8
        __builtin_prefetch(logits + 2 * (i + stride), 0, 0);

        const float l0 = logits[2 * i + 0];
        const float l1 = logits[2 * i + 1];
        const int   yi = y[i];
        const bool  mk = (mask[i] != 0);
        const int   wi = widx[i];

        // log-softmax (2 classes), same max-shift as jax.nn.log_softmax
        const float mx  = fmaxf(l0, l1);
        const float lse = mx + logf(expf(l0 - mx) + expf(l1 - mx));
        const float ly  = (yi == 1) ? l1 : l0;
        const float nll = lse - ly;
        const float w   = ((yi == 1) ? cw1 : cw0) * (mk ? 1.0f : 0.0f);
        a_wnll += (double)(w * nll);
        a_w    += (double)w;

        const bool  valid = mk && (wi >= 0);
        const float d_obs = fmaxf(xraw[8 * i + 2], 0.0f);
        keys[i] = valid ? __float_as_uint(d_obs) : 0xFFFFFFFFu;  // sentinel sorts last

        if (valid) {
            const float ap   = expf(l1 - lse);                 // softmax[:,1]
            const float rate = fmaxf(xraw[8 * i + 3], 0.0f);
            const float asz  = fmaxf(xraw[8 * i + 4], 0.0f) * 1000.0f;
            const float br   = rate * asz;
            int seg = wi; if (seg > NWIN - 1) seg = NWIN - 1;
            atomicAdd(&wsum[0 * NWIN + seg], ap);
            atomicAdd(&wsum[1 * NWIN + seg], ap * br);
            atomicAdd(&wsum[2 * NWIN + seg], ap * d_obs);
            atomicAdd(&wsum[3 * NWIN + seg], 1.0f);
        }
    }

    // wave32 shuffle reduction of the f64 CE partials, one atomic per wave
    for (int off = 16; off > 0; off >>= 1) {
        a_wnll += __shfl_down(a_wnll, off, 32);
        a_w    += __shfl_down(a_w,    off, 32);
    }
    if ((threadIdx.x & 31) == 0) {
        atomicAdd(&dacc[0], a_wnll);
        atomicAdd(&dacc[1], a_w);
    }

    __syncthreads();
    float* mypart = partials + (size_t)blockIdx.x * (4 * NWIN);
    for (int j = threadIdx.x; j < 4 * NWIN; j += blockDim.x) mypart[j] = wsum[j];
}

// ---------------------------------------------------------------------------
// 11-bit first-level histogram of quantile keys.
// Keys are double-buffer staged into LDS with global_load_async_to_lds_b128;
// each wave reads back only the LDS region it loaded, so only ASYNCcnt waits
// (no barriers) are needed. Async loads complete in order per the ISA.
// ---------------------------------------------------------------------------
__global__ __launch_bounds__(256) void hist0_kernel(const unsigned* __restrict__ keys,
                                                    unsigned* __restrict__ h0g) {
    __shared__ unsigned h[2048];
    __shared__ unsigned stage[2][TILE_KEYS];
    const int tid = threadIdx.x;
    for (int j = tid; j < 2048; j += 256) h[j] = 0u;
    __syncthreads();

    const int NT = N_ROWS / TILE_KEYS;   // 3906 full tiles (+256 tail keys)
    const unsigned lds0 = lds_off_u32(&stage[0][tid * 4]);
    const unsigned lds1 = lds_off_u32(&stage[1][tid * 4]);

    int tile = blockIdx.x;
    if (tile < NT) async_ld_b128(lds0, keys + (size_t)tile * TILE_KEYS + tid * 4);
    int cur = 0;
    for (; tile < NT; tile += gridDim.x) {
        const int nxt = tile + gridDim.x;
        if (nxt < NT) {
            async_ld_b128(cur ? lds0 : lds1, keys + (size_t)nxt * TILE_KEYS + tid * 4);
            wait_async_le1();
        } else {
            wait_async_le0();
        }
        #pragma unroll
        for (int e = 0; e < 4; ++e) {
            const unsigned k = stage[cur][tid * 4 + e];
            atomicAdd(&h[k >> 21], 1u);
        }
        cur ^= 1;
    }
    // tail (last partial tile), direct loads
    for (int i = NT * TILE_KEYS + blockIdx.x * 256 + tid; i < N_ROWS; i += gridDim.x * 256) {
        atomicAdd(&h[keys[i] >> 21], 1u);
    }
    __syncthreads();
    for (int j = tid; j < 2048; j += 256) {
        const unsigned v = h[j];
        if (v) atomicAdd(&h0g[j], v);
    }
}

// ---------------------------------------------------------------------------
// Reduce per-block window partials -> finals[4][NWIN] with WMMA:
//   finals = ones(1x256) x partials(256x16384), done as 64 chained
//   V_WMMA_F32_16X16X4_F32 ops per wave (A = all-ones, K-slot-permutation
//   invariant). D row M=0 sits in acc[0], lanes 0..15.
// ---------------------------------------------------------------------------
__global__ __launch_bounds__(256) void reduce_kernel(const float* __restrict__ partials,
                                                     float* __restrict__ finals) {
    const int wave = (blockIdx.x * blockDim.x + threadIdx.x) >> 5;
    const int lane = threadIdx.x & 31;
    const int out0 = wave * 16;
    if (out0 >= 4 * NWIN) return;            // wave-uniform
    const int n    = lane & 15;
    const int half = lane >> 4;

    v2f a; a.x = 1.0f; a.y = 1.0f;           // A = 16x4 ones
    v8f acc = {};
    for (int p0 = 0; p0 < PBLK; p0 += 4) {
        v2f b;                                // B = 4x16 tile of partials
        b.x = partials[(size_t)(p0 + (half ? 2 : 0)) * (4 * NWIN) + out0 + n];
        b.y = partials[(size_t)(p0 + (half ? 3 : 1)) * (4 * NWIN) + out0 + n];
        acc = __builtin_amdgcn_wmma_f32_16x16x4_f32(false, a, false, b,
                                                    (short)0, acc, false, false);
    }
    if (lane < 16) finals[out0 + n] = acc[0];
}

// ---------------------------------------------------------------------------
// pick0: n_valid, quantile position, first-level bin selection for both ranks.
// ---------------------------------------------------------------------------
__global__ void pick0_kernel(const float* __restrict__ finals,
                             const unsigned* __restrict__ h0g, SelState* st) {
    __shared__ float red[256];
    float s = 0.f;
    for (int j = threadIdx.x; j < NWIN; j += 256) s += finals[3 * NWIN + j];
    red[threadIdx.x] = s;
    __syncthreads();
    if (threadIdx.x == 0) {
        float nv = 0.f;
        for (int t = 0; t < 256; ++t) nv += red[t];   // integer-valued, exact in f32
        st->n_valid = nv;
        // mirror reference f32 arithmetic exactly
        const float pos = 0.75f * (nv - 1.0f);
        const float pf  = floorf(pos);
        st->frac = pos - pf;
        long lo = (long)pf;         if (lo < 0) lo = 0; if (lo > N_ROWS - 1) lo = N_ROWS - 1;
        long hi = (long)ceilf(pos); if (hi < 0) hi = 0; if (hi > N_ROWS - 1) hi = N_ROWS - 1;
        unsigned long rk[2] = { (unsigned long)lo, (unsigned long)hi };
        int done[2] = { 0, 0 };
        unsigned long cum = 0;
        for (int b = 0; b < 2048; ++b) {
            const unsigned long c = (unsigned long)h0g[b];
            for (int k = 0; k < 2; ++k) {
                if (!done[k] && rk[k] < cum + c) {
                    st->prefix[k] = (unsigned)b << 21;
                    st->rank[k]   = (unsigned)(rk[k] - cum);
                    done[k] = 1;
                }
            }
            cum += c;
        }
        for (int k = 0; k < 2; ++k)
            if (!done[k]) { st->prefix[k] = 2047u << 21; st->rank[k] = 0u; }
    }
}

// ---------------------------------------------------------------------------
// Radix-select refine: histogram the next field for keys matching each prefix,
// with the same async-to-LDS double-buffered staging as hist0.
// ---------------------------------------------------------------------------
__global__ __launch_bounds__(256) void refine_kernel(const unsigned* __restrict__ keys,
                                                     const SelState* __restrict__ st,
                                                     unsigned* __restrict__ rhg,
                                                     int shift, int bits) {
    __shared__ unsigned rh[512];
    __shared__ unsigned stage[2][TILE_KEYS];
    const int tid = threadIdx.x;
    for (int j = tid; j < 512; j += 256) rh[j] = 0u;
    __syncthreads();

    const unsigned top   = (unsigned)(shift + bits);
    const unsigned p0    = st->prefix[0] >> top;
    const unsigned p1    = st->prefix[1] >> top;
    const unsigned bmask = (1u << bits) - 1u;

    const int NT = N_ROWS / TILE_KEYS;
    const unsigned lds0 = lds_off_u32(&stage[0][tid * 4]);
    const unsigned lds1 = lds_off_u32(&stage[1][tid * 4]);

    int tile = blockIdx.x;
    if (tile < NT) async_ld_b128(lds0, keys + (size_t)tile * TILE_KEYS + tid * 4);
    int cur = 0;
    for (; tile < NT; tile += gridDim.x) {
        const int nxt = tile + gridDim.x;
        if (nxt < NT) {
            async_ld_b128(cur ? lds0 : lds1, keys + (size_t)nxt * TILE_KEYS + tid * 4);
            wait_async_le1();
        } else {
            wait_async_le0();
        }
        #pragma unroll
        for (int e = 0; e < 4; ++e) {
            const unsigned k   = stage[cur][tid * 4 + e];
            const unsigned kt  = k >> top;
            const unsigned bin = (k >> shift) & bmask;
            if (kt == p0) atomicAdd(&rh[bin], 1u);
            if (kt == p1) atomicAdd(&rh[256 + bin], 1u);
        }
        cur ^= 1;
    }
    for (int i = NT * TILE_KEYS + blockIdx.x * 256 + tid; i < N_ROWS; i += gridDim.x * 256) {
        const unsigned k   = keys[i];
        const unsigned kt  = k >> top;
        const unsigned bin = (k >> shift) & bmask;
        if (kt == p0) atomicAdd(&rh[bin], 1u);
        if (kt == p1) atomicAdd(&rh[256 + bin], 1u);
    }
    __syncthreads();
    for (int j = tid; j < 512; j += 256) {
        const unsigned v = rh[j];
        if (v) atomicAdd(&rhg[j], v);
    }
}

__global__ void pick_kernel(unsigned* __restrict__ rhg, SelState* st,
                            int shift, int bits, int last) {
    if (threadIdx.x == 0 && blockIdx.x == 0) {
        const int nb = 1 << bits;
        for (int k = 0; k < 2; ++k) {
            const unsigned rk = st->rank[k];
            unsigned cum = 0, res = 0;
            int sel = nb - 1;
            for (int b = 0; b < nb; ++b) {
                const unsigned c = rhg[k * 256 + b];
                if (rk < cum + c) { sel = b; res = rk - cum; break; }
                cum += c;
            }
            st->prefix[k] |= (unsigned)sel << shift;
            st->rank[k] = res;
        }
        for (int j = 0; j < 512; ++j) rhg[j] = 0u;   // ready for next level
        if (last) {
            const float x0 = __uint_as_float(st->prefix[0]);
            const float x1 = __uint_as_float(st->prefix[1]);
            st->xs0 = x0; st->xs1 = x1;
            const float f = st->frac;
            st->ref_dobs = fmaxf(x0 * (1.0f - f) + x1 * f, 1e-6f);
        }
    }
}

// ---------------------------------------------------------------------------
// Final: parallel max-plus scan of the queue recurrence (f64) + loss assembly.
//   q_t = max(0, q_{t-1} + b_t)  ==  S_t - min_{0<=j<=t} S_j
// ---------------------------------------------------------------------------
__global__ void final_kernel(const float* __restrict__ finals, const SelState* __restrict__ st,
                             const double* __restrict__ dacc, float* __restrict__ out) {
    __shared__ double sA[256];
    __shared__ double sB[256];
    __shared__ double sC[256];
    const int tid = threadIdx.x;
    const float refd = st->ref_dobs;
    const int W0 = tid * (NWIN / 256);   // 16 windows per thread

    // pass 1: per-segment sums of b
    double bloc[NWIN / 256];
    double segsum = 0.0;
    for (int j = 0; j < NWIN / 256; ++j) {
        const int wdx = W0 + j;
        const float sp  = finals[0 * NWIN + wdx];
        const float spr = finals[1 * NWIN + wdx];
        const float cnt = finals[3 * NWIN + wdx];
        const bool inc = (cnt >= 1.0f) && (sp >= 1e-8f);
        const double b = inc ? ((double)(spr * DELTA_T) - (double)SERVICE) : 0.0;
        bloc[j] = b;
        segsum += b;
    }
    sA[tid] = segsum;
    __syncthreads();
    if (tid == 0) {                       // exclusive prefix-sum over 256 segments
        double run = 0.0;
        for (int t = 0; t < 256; ++t) { const double v = sA[t]; sA[t] = run; run += v; }
    }
    __syncthreads();
    const double base = sA[tid];

    // local inclusive-S minimum within segment
    double S = base, mloc = 1e300;
    for (int j = 0; j < NWIN / 256; ++j) { S += bloc[j]; if (S < mloc) mloc = S; }
    sB[tid] = mloc;
    __syncthreads();
    if (tid == 0) {                       // exclusive prefix-min (seeded with S_0 = 0)
        double run = 0.0;
        for (int t = 0; t < 256; ++t) { const double v = sB[t]; sB[t] = run; run = fmin(run, v); }
    }
    __syncthreads();
    const double minbase = sB[tid];
    __syncthreads();                      // sA/sB reused below

    // pass 2: per-window queue + contributions
    double fsum = 0.0, lsum = 0.0, isum = 0.0;
    S = base;
    double mrun = minbase;
    for (int j = 0; j < NWIN / 256; ++j) {
        const int wdx = W0 + j;
        const float sp  = finals[0 * NWIN + wdx];
        const float spr = finals[1 * NWIN + wdx];
        const float spd = finals[2 * NWIN + wdx];
        const float cnt = finals[3 * NWIN + wdx];
        const bool inc = (cnt >= 1.0f) && (sp >= 1e-8f);
        S += bloc[j];
        if (S < mrun) mrun = S;
        if (inc) {
            const float qn       = (float)(S - mrun);             // queue_next
            const float d_mean   = spd / (sp + 1e-6f);
            const float d_scaled = d_mean / refd;
            const float obs      = fmaxf(d_scaled - 1.0f, 0.0f) * SERVICE;
            float fl = (qn - obs) / (SERVICE + 1e-6f);
            fl = fl * fl;
            float rho = spr / (CAPACITY + 1e-6f);
            rho = fminf(fmaxf(rho, 0.0f), 0.995f);
            const float d_theory = 1.0f / (1.0f - rho + 1e-6f);
            const float lat = fmaxf(d_theory - d_scaled, 0.0f);
            fsum += (double)fl; lsum += (double)lat; isum += 1.0;
        }
    }
    sA[tid] = fsum; sB[tid] = lsum; sC[tid] = isum;
    __syncthreads();
    if (tid == 0) {
        double F = 0, L = 0, I = 0;
        for (int t = 0; t < 256; ++t) { F += sA[t]; L += sB[t]; I += sC[t]; }
        const float n_inc = (float)I;
        const float l_flow = (n_inc > 0.f) ? (float)(F / (double)fmaxf(n_inc, 1.0f)) : 0.0f;
        const float l_lat  = (n_inc > 0.f) ? (float)(L / (double)fmaxf(n_inc, 1.0f)) : 0.0f;
        const float l_data = (float)(dacc[0] / dacc[1]);
        const float l_total = l_data + ALPHA_C * l_flow + BETA_C * l_lat;
        out[0] = l_total; out[1] = l_data; out[2] = l_flow; out[3] = l_lat;
    }
}

// ---------------------------------------------------------------------------
extern "C" void kernel_launch(void* const* d_in, const int* in_sizes, int n_in,
                              void* d_out, int out_size, void* d_ws, size_t ws_size,
                              hipStream_t stream) {
    (void)in_sizes; (void)n_in; (void)out_size; (void)ws_size;

    const float* logits = (const float*)d_in[0];
    const int*   y      = (const int*)d_in[1];
    const int*   mask   = (const int*)d_in[2];     // jnp bool staged as int32 per harness rules
    const float* xraw   = (const float*)d_in[3];
    const int*   widx   = (const int*)d_in[4];
    const float* cw     = (const float*)d_in[5];
    float* out = (float*)d_out;

    char* ws = (char*)d_ws;
    size_t off = 0;
    auto take = [&](size_t bytes) {
        off = (off + 255) & ~(size_t)255;
        size_t r = off; off += bytes; return r;
    };
    unsigned* keys     = (unsigned*)(ws + take((size_t)N_ROWS * 4));          // 16 MB
    float*    partials = (float*)   (ws + take((size_t)PBLK * 4 * NWIN * 4)); // 16 MB
    float*    finals   = (float*)   (ws + take((size_t)4 * NWIN * 4));        // 64 KB
    unsigned* h0g      = (unsigned*)(ws + take(2048 * 4));
    unsigned* rhg      = (unsigned*)(ws + take(512 * 4));
    SelState* st       = (SelState*)(ws + take(sizeof(SelState)));
    double*   dacc     = (double*)  (ws + take(2 * sizeof(double)));

    zero_kernel  <<<1,   256, 0, stream>>>(h0g, rhg, st, dacc);
    main_kernel  <<<PBLK, TMAIN, 0, stream>>>(logits, y, mask, xraw, widx, cw,
                                              keys, partials, dacc);
    hist0_kernel <<<256, 256, 0, stream>>>(keys, h0g);
    reduce_kernel<<<128, 256, 0, stream>>>(partials, finals);   // 1024 waves x 16 outputs
    pick0_kernel <<<1,   256, 0, stream>>>(finals, h0g, st);

    refine_kernel<<<256, 256, 0, stream>>>(keys, st, rhg, 13, 8);
    pick_kernel  <<<1,   1,   0, stream>>>(rhg, st, 13, 8, 0);
    refine_kernel<<<256, 256, 0, stream>>>(keys, st, rhg, 5, 8);
    pick_kernel  <<<1,   1,   0, stream>>>(rhg, st, 5, 8, 0);
    refine_kernel<<<256, 256, 0, stream>>>(keys, st, rhg, 0, 5);
    pick_kernel  <<<1,   1,   0, stream>>>(rhg, st, 0, 5, 1);

    final_kernel <<<1,   256, 0, stream>>>(finals, st, dacc, out);
}